// HGCNEncoder_75453985456636
// MI455X (gfx1250) — compile-verified
//
#include <hip/hip_runtime.h>
#include <hip/hip_bf16.h>
#include <cmath>

#define B_ 4
#define N_ 8192
#define E_ 2048
#define S_ 64
#define D_ 16

typedef __attribute__((ext_vector_type(2))) float v2f;
typedef __attribute__((ext_vector_type(4))) float v4f;
typedef __attribute__((ext_vector_type(8))) float v8f;

// ---------------------------------------------------------------------------
// k1: abs_w[b,e] = | gelu(states @ W1^T + b1) @ W2^T + b2 |   (exact erf GELU)
// One block per batch; tiny.
// ---------------------------------------------------------------------------
__global__ void __launch_bounds__(256)
k1_absw(const float* __restrict__ states,
        const float* __restrict__ W1, const float* __restrict__ b1,
        const float* __restrict__ W2, const float* __restrict__ b2,
        float* __restrict__ absw) {
    __shared__ float ss[S_];
    __shared__ float sh[S_];
    const int b = blockIdx.x, t = threadIdx.x;
    if (t < S_) ss[t] = states[b * S_ + t];
    __syncthreads();
    if (t < S_) {
        float acc = b1[t];
#pragma unroll
        for (int k = 0; k < S_; ++k) acc += ss[k] * W1[t * S_ + k];
        // exact GELU: 0.5*x*(1+erf(x/sqrt(2)))
        sh[t] = 0.5f * acc * (1.0f + erff(acc * 0.70710678118654752f));
    }
    __syncthreads();
    for (int e = t; e < E_; e += 256) {
        float acc = b2[e];
#pragma unroll
        for (int s = 0; s < S_; ++s) acc += sh[s] * W2[e * S_ + s];
        absw[b * E_ + e] = fabsf(acc);
    }
}

// ---------------------------------------------------------------------------
// k2: pass 1 over H (bandwidth-bound).
//   d[b,n]  = sum_e H[b,n,e] * abs_w[b,e]
//   nfp[b,n,d] = (d>0 ? rsqrt(d) : 0) * nf[b,n,d]
// One wave per (b,n) row; 8 waves/block; abs_w for the batch staged in LDS.
// ---------------------------------------------------------------------------
__global__ void __launch_bounds__(256)
k2_deg(const float* __restrict__ H, const float* __restrict__ nf,
       const float* __restrict__ absw, float* __restrict__ nfp) {
    __shared__ float sw[E_];  // 8 KB
    const int tid  = threadIdx.x;
    const int wave = tid >> 5, lane = tid & 31;
    const int row  = blockIdx.x * 8 + wave;   // rows of a block share one batch
    const int b    = row / N_;
    for (int i = tid; i < E_; i += 256) sw[i] = absw[b * E_ + i];
    __syncthreads();

    const float* hrow = H + (size_t)row * E_;
    float acc = 0.f;
#pragma unroll
    for (int i = 0; i < 16; ++i) {              // 16 * 32 lanes * 4 = 2048
        const int idx = i * 128 + lane * 4;
        v4f h = *(const v4f*)(hrow + idx);      // global_load_b128
        acc += h.x * sw[idx] + h.y * sw[idx + 1] + h.z * sw[idx + 2] + h.w * sw[idx + 3];
    }
#pragma unroll
    for (int off = 16; off > 0; off >>= 1) acc += __shfl_xor(acc, off, 32);
    const float dinv = acc > 0.f ? rsqrtf(acc) : 0.f;
    if (lane < D_)
        nfp[(size_t)row * D_ + lane] = dinv * nf[(size_t)row * D_ + lane];
}

// ---------------------------------------------------------------------------
// k3: pass 2 over H with fp32 WMMA.
//   X[b, e, :] = sum_n H[b,n,e] * nfp[b,n,:]   (A = H^T tile, B = nfp rows)
//   out = leaky_relu( abs_w[b,e] * X , 0.1 )
// One block per (b, 32-wide e strip): each wave covers TWO 16-wide e-tiles
// (shared B operand -> 3 VMEM loads per wmma instead of 4), and 8 waves split
// the K (=N) dimension; partials reduced through LDS.
// A-matrix (16x4 f32) layout per ISA: lane m=lane%16 -> M (e-col),
// kb=(lane/16)*2 -> K base; VGPR0=K kb, VGPR1=K kb+1. B mirrors with N=lane%16.
// ---------------------------------------------------------------------------
#define KCHUNK (N_ / 8)   // 1024 n-values per wave
#define PFDIST 128        // prefetch distance in n-rows (~1 MB ahead per stream)

__global__ void __launch_bounds__(256)
k3_gemm(const float* __restrict__ H, const float* __restrict__ nfp,
        const float* __restrict__ absw, float* __restrict__ out) {
    __shared__ float red[8][512];               // 16 KB
    const int b    = blockIdx.x >> 6;           // E_/32 = 64 strips per batch
    const int e0   = (blockIdx.x & 63) << 5;    // 32-wide e strip
    const int tid  = threadIdx.x;
    const int wave = tid >> 5, lane = tid & 31;
    const int m    = lane & 15;                 // e-col (A) / d-col (B)
    const int kb   = (lane >> 4) << 1;          // K sub-base per lane half

    const float* Hp0 = H   + (size_t)b * N_ * E_ + (size_t)(e0 + m);        // tile 0
    const float* Hp1 = Hp0 + 16;                                            // tile 1
    const float* Pp  = nfp + (size_t)b * N_ * D_ + (size_t)m;

    v8f c0 = {};
    v8f c1 = {};
    const int n_begin = wave * KCHUNK;
    for (int n0 = n_begin; n0 < n_begin + KCHUNK; n0 += 16) {
        // keep the HBM pipe primed (global_prefetch_b8, speculative)
        __builtin_prefetch(&Hp0[(size_t)(n0 + PFDIST) * E_], 0, 0);
        __builtin_prefetch(&Hp1[(size_t)(n0 + PFDIST) * E_], 0, 0);
#pragma unroll
        for (int k = 0; k < 4; ++k) {
            const int nA = n0 + 4 * k + kb;
            v2f a0, a1, bv;
            a0.x = Hp0[(size_t)nA * E_];        // H[n, e0+m]
            a0.y = Hp0[(size_t)(nA + 1) * E_];
            a1.x = Hp1[(size_t)nA * E_];        // H[n, e0+16+m]
            a1.y = Hp1[(size_t)(nA + 1) * E_];
            bv.x = Pp[(size_t)nA * D_];         // nfp[n, m]  (shared by both tiles)
            bv.y = Pp[(size_t)(nA + 1) * D_];
            c0 = __builtin_amdgcn_wmma_f32_16x16x4_f32(
                    false, a0, false, bv, (short)0, c0, false, false);
            c1 = __builtin_amdgcn_wmma_f32_16x16x4_f32(
                    false, a1, false, bv, (short)0, c1, false, false);
        }
    }

    // C/D layout: VGPR r -> M = r + 8*(lane/16), N = lane%16
#pragma unroll
    for (int r = 0; r < 8; ++r) {
        const int pos = ((r + ((lane >> 4) << 3)) << 4) + m;
        red[wave][pos]       = c0[r];
        red[wave][256 + pos] = c1[r];
    }
    __syncthreads();

#pragma unroll
    for (int q = 0; q < 2; ++q) {               // 256 threads cover 512 elems
        const int idx = tid + q * 256;
        float s = 0.f;
#pragma unroll
        for (int w = 0; w < 8; ++w) s += red[w][idx];
        const int me = idx >> 4, dd = idx & 15; // me = e within strip (0..31)
        float v = s * absw[b * E_ + e0 + me];
        v = v > 0.f ? v : 0.1f * v;             // leaky_relu slope 0.1
        out[((size_t)b * E_ + (size_t)(e0 + me)) * D_ + dd] = v;
    }
}

// ---------------------------------------------------------------------------
extern "C" void kernel_launch(void* const* d_in, const int* in_sizes, int n_in,
                              void* d_out, int out_size, void* d_ws, size_t ws_size,
                              hipStream_t stream) {
    const float* nf     = (const float*)d_in[0];  // (B,N,D)
    const float* H      = (const float*)d_in[1];  // (B,N,E)
    const float* states = (const float*)d_in[2];  // (B,S)
    const float* W1     = (const float*)d_in[3];  // (S,S)
    const float* b1     = (const float*)d_in[4];  // (S,)
    const float* W2     = (const float*)d_in[5];  // (E,S)
    const float* b2     = (const float*)d_in[6];  // (E,)
    float* out = (float*)d_out;                   // (B,E,D)

    float* nfp  = (float*)d_ws;                   // B*N*D floats (2 MB)
    float* absw = nfp + (size_t)B_ * N_ * D_;     // B*E floats (32 KB)

    k1_absw<<<B_, 256, 0, stream>>>(states, W1, b1, W2, b2, absw);
    k2_deg <<<(B_ * N_) / 8, 256, 0, stream>>>(H, nf, absw, nfp);
    k3_gemm<<<B_ * (E_ / 32), 256, 0, stream>>>(H, nfp, absw, out);
}